// PointNetSegmenter_40192303956813
// MI455X (gfx1250) — compile-verified
//
#include <hip/hip_runtime.h>
#include <hip/hip_bf16.h>
#include <math.h>

#define N_TOTAL   262144
#define NSAMP     1024
#define KNN_K     16
#define NCLS      13
#define HIDDEN    128
#define EPSV      1e-8f

typedef __attribute__((ext_vector_type(2))) float v2f;
typedef __attribute__((ext_vector_type(8))) float v8f;

// ---------------------------------------------------------------------------
// Kernel 1: farthest point sampling (single persistent workgroup, 32 waves).
// min_d2 lives in global ws (1 MB, L2-resident); argmax via LDS tree.
// ---------------------------------------------------------------------------
__global__ void __launch_bounds__(1024) fps_kernel(
    const float* __restrict__ pos, const float* __restrict__ col,
    float* __restrict__ min_d2, int* __restrict__ sample_idx,
    float* __restrict__ dxyz, float* __restrict__ drgb)
{
    __shared__ float s_val[1024];
    __shared__ int   s_idx[1024];
    __shared__ int   s_samp[NSAMP];
    __shared__ int   s_last;

    const int tid = threadIdx.x;

    for (int i = tid; i < N_TOTAL; i += 1024) min_d2[i] = 1e10f;
    if (tid == 0) { s_last = 0; s_samp[0] = 0; sample_idx[0] = 0; }
    __syncthreads();

    for (int s = 1; s < NSAMP; ++s) {
        const int last = s_last;
        const float px = pos[last * 3 + 0];
        const float py = pos[last * 3 + 1];
        const float pz = pos[last * 3 + 2];

        float bestv = -1.0f;
        int   besti = 0;
        for (int i = tid; i < N_TOTAL; i += 1024) {
            const float dx = pos[i * 3 + 0] - px;
            const float dy = pos[i * 3 + 1] - py;
            const float dz = pos[i * 3 + 2] - pz;
            const float d2 = dx * dx + dy * dy + dz * dz;
            float m = min_d2[i];
            m = fminf(m, d2);
            min_d2[i] = m;
            if (m > bestv) { bestv = m; besti = i; }   // strict >: first-index tie-break
        }
        s_val[tid] = bestv;
        s_idx[tid] = besti;
        __syncthreads();

        for (int off = 512; off > 0; off >>= 1) {
            if (tid < off) {
                const float v2 = s_val[tid + off];
                const int   i2 = s_idx[tid + off];
                if (v2 > s_val[tid] || (v2 == s_val[tid] && i2 < s_idx[tid])) {
                    s_val[tid] = v2; s_idx[tid] = i2;
                }
            }
            __syncthreads();
        }
        if (tid == 0) {
            s_last = s_idx[0];
            s_samp[s] = s_idx[0];
            sample_idx[s] = s_idx[0];
        }
        __syncthreads();
    }

    // gather sampled xyz / rgb (one thread per sample; indices from LDS)
    {
        const int j = s_samp[tid];
        dxyz[tid * 3 + 0] = pos[j * 3 + 0];
        dxyz[tid * 3 + 1] = pos[j * 3 + 1];
        dxyz[tid * 3 + 2] = pos[j * 3 + 2];
        drgb[tid * 3 + 0] = col[j * 3 + 0];
        drgb[tid * 3 + 1] = col[j * 3 + 1];
        drgb[tid * 3 + 2] = col[j * 3 + 2];
    }
}

// ---------------------------------------------------------------------------
// Kernel 2: kNN covariance normals (closed-form 3x3 symmetric eigensolver).
// One thread per sample; all 1024 sample coords resident in LDS (12 KB).
// ---------------------------------------------------------------------------
__device__ void smallest_eigvec3(float a00, float a01, float a02,
                                 float a11, float a12, float a22,
                                 float& vx, float& vy, float& vz)
{
    const float p1 = a01 * a01 + a02 * a02 + a12 * a12;
    float lmin;
    if (p1 < 1e-30f) {
        lmin = fminf(a00, fminf(a11, a22));
    } else {
        const float q  = (a00 + a11 + a22) * (1.0f / 3.0f);
        const float p2 = (a00 - q) * (a00 - q) + (a11 - q) * (a11 - q) +
                         (a22 - q) * (a22 - q) + 2.0f * p1;
        const float p  = sqrtf(p2 * (1.0f / 6.0f));
        const float ip = 1.0f / p;
        const float b00 = (a00 - q) * ip, b01 = a01 * ip, b02 = a02 * ip;
        const float b11 = (a11 - q) * ip, b12 = a12 * ip;
        const float b22 = (a22 - q) * ip;
        float r = 0.5f * (b00 * (b11 * b22 - b12 * b12)
                        - b01 * (b01 * b22 - b12 * b02)
                        + b02 * (b01 * b12 - b11 * b02));
        r = fmaxf(-1.0f, fminf(1.0f, r));
        const float phi = acosf(r) * (1.0f / 3.0f);
        lmin = q + 2.0f * p * cosf(phi + 2.0943951023931953f);  // smallest eigenvalue
    }
    // eigenvector from cross products of rows of (A - lmin I)
    const float r0x = a00 - lmin, r0y = a01,        r0z = a02;
    const float r1x = a01,        r1y = a11 - lmin, r1z = a12;
    const float r2x = a02,        r2y = a12,        r2z = a22 - lmin;

    float c0x = r0y * r1z - r0z * r1y, c0y = r0z * r1x - r0x * r1z, c0z = r0x * r1y - r0y * r1x;
    float c1x = r0y * r2z - r0z * r2y, c1y = r0z * r2x - r0x * r2z, c1z = r0x * r2y - r0y * r2x;
    float c2x = r1y * r2z - r1z * r2y, c2y = r1z * r2x - r1x * r2z, c2z = r1x * r2y - r1y * r2x;

    const float n0 = c0x * c0x + c0y * c0y + c0z * c0z;
    const float n1 = c1x * c1x + c1y * c1y + c1z * c1z;
    const float n2 = c2x * c2x + c2y * c2y + c2z * c2z;

    float bx = c0x, by = c0y, bz = c0z, bn = n0;
    if (n1 > bn) { bx = c1x; by = c1y; bz = c1z; bn = n1; }
    if (n2 > bn) { bx = c2x; by = c2y; bz = c2z; bn = n2; }
    if (bn < 1e-30f) { bx = 1.0f; by = 0.0f; bz = 0.0f; bn = 1.0f; }
    const float inv = 1.0f / (sqrtf(bn) + EPSV);
    vx = bx * inv; vy = by * inv; vz = bz * inv;
}

__global__ void __launch_bounds__(256) normals_kernel(
    const float* __restrict__ dxyz, float* __restrict__ dnrm)
{
    __shared__ float sx[NSAMP], sy[NSAMP], sz[NSAMP];
    const int tid = threadIdx.x;
    for (int i = tid; i < NSAMP; i += 256) {
        sx[i] = dxyz[i * 3 + 0];
        sy[i] = dxyz[i * 3 + 1];
        sz[i] = dxyz[i * 3 + 2];
    }
    __syncthreads();

    const int i = blockIdx.x * 256 + tid;
    const float xi = sx[i], yi = sy[i], zi = sz[i];

    // (k+1)-NN selection by repeated lexicographic-min passes over (d2, j);
    // pass 0 picks self (d2 = 0) and is excluded from the accumulators.
    float prev_d = -1.0f; int prev_j = -1;
    float S1x = 0.f, S1y = 0.f, S1z = 0.f;
    float Sxx = 0.f, Sxy = 0.f, Sxz = 0.f, Syy = 0.f, Syz = 0.f, Szz = 0.f;

    for (int k = 0; k < KNN_K + 1; ++k) {
        float bd = 3.4e38f; int bj = 0;
        for (int j = 0; j < NSAMP; ++j) {
            const float dx = sx[j] - xi, dy = sy[j] - yi, dz = sz[j] - zi;
            const float d2 = dx * dx + dy * dy + dz * dz;
            const bool after = (d2 > prev_d) || (d2 == prev_d && j > prev_j);
            if (after && d2 < bd) { bd = d2; bj = j; }
        }
        prev_d = bd; prev_j = bj;
        if (k > 0) {
            const float nx = sx[bj], ny = sy[bj], nz = sz[bj];
            S1x += nx; S1y += ny; S1z += nz;
            Sxx += nx * nx; Sxy += nx * ny; Sxz += nx * nz;
            Syy += ny * ny; Syz += ny * nz; Szz += nz * nz;
        }
    }

    const float inv_k = 1.0f / (float)KNN_K;
    const float mx = S1x * inv_k, my = S1y * inv_k, mz = S1z * inv_k;
    const float inv_km1 = 1.0f / (float)(KNN_K - 1);
    const float cxx = (Sxx - (float)KNN_K * mx * mx) * inv_km1;
    const float cxy = (Sxy - (float)KNN_K * mx * my) * inv_km1;
    const float cxz = (Sxz - (float)KNN_K * mx * mz) * inv_km1;
    const float cyy = (Syy - (float)KNN_K * my * my) * inv_km1;
    const float cyz = (Syz - (float)KNN_K * my * mz) * inv_km1;
    const float czz = (Szz - (float)KNN_K * mz * mz) * inv_km1;

    float vx, vy, vz;
    smallest_eigvec3(cxx, cxy, cxz, cyy, cyz, czz, vx, vy, vz);
    dnrm[i * 3 + 0] = vx;
    dnrm[i * 3 + 1] = vy;
    dnrm[i * 3 + 2] = vz;
}

// ---------------------------------------------------------------------------
// Kernel 3: build zero-padded WMMA operands:
//   feats_pad : 1024 x 12  = [xyz, rgb, nrm, 0, 0, 0]
//   W1p       : 12 x 128   (rows 9..11 zero)
//   W3p       : 128 x 16   (cols 13..15 zero)
//   b3p       : 16         (elems 13..15 zero)
// ---------------------------------------------------------------------------
__global__ void __launch_bounds__(256) prep_kernel(
    const float* __restrict__ dxyz, const float* __restrict__ drgb,
    const float* __restrict__ dnrm, const float* __restrict__ W1,
    const float* __restrict__ W3, const float* __restrict__ b3,
    float* __restrict__ feats_pad, float* __restrict__ W1p,
    float* __restrict__ W3p, float* __restrict__ b3p)
{
    const int id = blockIdx.x * 256 + threadIdx.x;
    if (id < NSAMP * 12) {
        const int i = id / 12, c = id % 12;
        float v = 0.0f;
        if (c < 3)      v = dxyz[i * 3 + c];
        else if (c < 6) v = drgb[i * 3 + (c - 3)];
        else if (c < 9) v = dnrm[i * 3 + (c - 6)];
        feats_pad[id] = v;
        return;
    }
    const int id2 = id - NSAMP * 12;
    if (id2 < 12 * HIDDEN) {
        const int k = id2 / HIDDEN, j = id2 % HIDDEN;
        W1p[id2] = (k < 9) ? W1[k * HIDDEN + j] : 0.0f;
        return;
    }
    const int id3 = id2 - 12 * HIDDEN;
    if (id3 < HIDDEN * 16) {
        const int k = id3 / 16, c = id3 % 16;
        W3p[id3] = (c < NCLS) ? W3[k * NCLS + c] : 0.0f;
        return;
    }
    const int id4 = id3 - HIDDEN * 16;
    if (id4 < 16) {
        b3p[id4] = (id4 < NCLS) ? b3[id4] : 0.0f;
    }
}

// ---------------------------------------------------------------------------
// Kernel 4 (x3 instantiations): C = act(A @ B + bias) via V_WMMA_F32_16X16X4_F32
// A: 1024 x K (row-major), B: K x N (row-major), C: 1024 x N.
// One 16x16 output tile per wave; K fully unrolled at compile time.
// A-frag: lane = M%16, VGPR pair = K+{0,1} (lanes 0-15) / K+{2,3} (lanes 16-31).
// B-frag: same K split, row-striped across lanes.  C/D: ISA 16x16 f32 layout.
// ---------------------------------------------------------------------------
template<int K, int N, int NTILE_N, bool RELU>
__global__ void __launch_bounds__(128) wmma_gemm_kernel(
    const float* __restrict__ A, const float* __restrict__ B,
    const float* __restrict__ bias, float* __restrict__ C)
{
    const int wave = (blockIdx.x * 128 + threadIdx.x) >> 5;
    const int lane = threadIdx.x & 31;
    const int mt = wave / NTILE_N;
    const int nt = wave - mt * NTILE_N;
    const int mrow  = mt * 16 + (lane & 15);
    const int ncol  = nt * 16 + (lane & 15);
    const int khalf = (lane >> 4) * 2;   // 0 or 2

    const float* ap = A + mrow * K + khalf;
    const float* bp = B + khalf * N + ncol;

    v8f c = {};
    #pragma unroll
    for (int k = 0; k < K; k += 4) {
        v2f a, b;
        a.x = ap[0];
        a.y = ap[1];
        b.x = bp[0];
        b.y = bp[N];
        c = __builtin_amdgcn_wmma_f32_16x16x4_f32(
                /*neg_a=*/false, a, /*neg_b=*/false, b,
                /*c_mod=*/(short)0, c, /*reuse_a=*/false, /*reuse_b=*/false);
        ap += 4;
        bp += 4 * N;
    }

    const int rbase = mt * 16 + 8 * (lane >> 4);
    const float bv = bias[ncol];
    #pragma unroll
    for (int v = 0; v < 8; ++v) {
        float val = c[v] + bv;
        if (RELU) val = fmaxf(val, 0.0f);
        C[(rbase + v) * N + ncol] = val;
    }
}

// ---------------------------------------------------------------------------
// Kernel 5: softmax + argmax per sample row (reads 1024 x 16 padded logits).
// ---------------------------------------------------------------------------
__global__ void __launch_bounds__(256) softmax_kernel(
    const float* __restrict__ pred_pad, float* __restrict__ feat_down,
    float* __restrict__ lbl_down)
{
    const int i = blockIdx.x * 256 + threadIdx.x;   // 0..NSAMP-1
    float pred[NCLS];
    #pragma unroll
    for (int c = 0; c < NCLS; ++c) pred[c] = pred_pad[i * 16 + c];

    float mx = pred[0]; int am = 0;
    #pragma unroll
    for (int c = 1; c < NCLS; ++c)
        if (pred[c] > mx) { mx = pred[c]; am = c; }   // first-index tie-break
    float sum = 0.0f;
    #pragma unroll
    for (int c = 0; c < NCLS; ++c) { const float e = __expf(pred[c] - mx); pred[c] = e; sum += e; }
    const float inv = 1.0f / sum;
    #pragma unroll
    for (int c = 0; c < NCLS; ++c) feat_down[i * NCLS + c] = pred[c] * inv;
    lbl_down[i] = (float)am;
}

// ---------------------------------------------------------------------------
// Kernel 6: per-point nearest sample (argmin over 1024 LDS-resident samples)
// then gather softmax row + label into d_out.  Each thread prefetches its own
// position (global_prefetch_b8) while the block stages the sample set to LDS.
// ---------------------------------------------------------------------------
__global__ void __launch_bounds__(256) nearest_kernel(
    const float* __restrict__ pos, const float* __restrict__ dxyz,
    const float* __restrict__ feat_down, const float* __restrict__ lbl_down,
    float* __restrict__ out_feat, float* __restrict__ out_lbl)
{
    __shared__ float sx[NSAMP], sy[NSAMP], sz[NSAMP];
    const int tid = threadIdx.x;
    const int p = blockIdx.x * 256 + tid;

    __builtin_prefetch(&pos[p * 3], 0, 0);   // overlap HBM/L2 latency with LDS fill

    for (int i = tid; i < NSAMP; i += 256) {
        sx[i] = dxyz[i * 3 + 0];
        sy[i] = dxyz[i * 3 + 1];
        sz[i] = dxyz[i * 3 + 2];
    }
    __syncthreads();

    const float x = pos[p * 3 + 0], y = pos[p * 3 + 1], z = pos[p * 3 + 2];

    float bd = 3.4e38f; int bs = 0;
    #pragma unroll 4
    for (int s = 0; s < NSAMP; ++s) {
        const float dx = sx[s] - x, dy = sy[s] - y, dz = sz[s] - z;
        const float d2 = dx * dx + dy * dy + dz * dz;
        if (d2 < bd) { bd = d2; bs = s; }             // strict <: first-index argmin
    }
    #pragma unroll
    for (int c = 0; c < NCLS; ++c) out_feat[p * NCLS + c] = feat_down[bs * NCLS + c];
    out_lbl[p] = lbl_down[bs];
}

// ---------------------------------------------------------------------------
extern "C" void kernel_launch(void* const* d_in, const int* in_sizes, int n_in,
                              void* d_out, int out_size, void* d_ws, size_t ws_size,
                              hipStream_t stream)
{
    const float* pos = (const float*)d_in[0];
    const float* col = (const float*)d_in[1];
    const float* W1  = (const float*)d_in[2];
    const float* b1  = (const float*)d_in[3];
    const float* W2  = (const float*)d_in[4];
    const float* b2  = (const float*)d_in[5];
    const float* W3  = (const float*)d_in[6];
    const float* b3  = (const float*)d_in[7];

    char* ws = (char*)d_ws;
    float* min_d2     = (float*)ws;  ws += (size_t)N_TOTAL * 4;
    int*   sample_idx = (int*)ws;    ws += (size_t)NSAMP * 4;
    float* dxyz       = (float*)ws;  ws += (size_t)NSAMP * 3 * 4;
    float* drgb       = (float*)ws;  ws += (size_t)NSAMP * 3 * 4;
    float* dnrm       = (float*)ws;  ws += (size_t)NSAMP * 3 * 4;
    float* feats_pad  = (float*)ws;  ws += (size_t)NSAMP * 12 * 4;
    float* W1p        = (float*)ws;  ws += (size_t)12 * HIDDEN * 4;
    float* W3p        = (float*)ws;  ws += (size_t)HIDDEN * 16 * 4;
    float* b3p        = (float*)ws;  ws += (size_t)16 * 4;
    float* h1         = (float*)ws;  ws += (size_t)NSAMP * HIDDEN * 4;
    float* h2         = (float*)ws;  ws += (size_t)NSAMP * HIDDEN * 4;
    float* pred_pad   = (float*)ws;  ws += (size_t)NSAMP * 16 * 4;
    float* feat_down  = (float*)ws;  ws += (size_t)NSAMP * NCLS * 4;
    float* lbl_down   = (float*)ws;  ws += (size_t)NSAMP * 4;

    float* out_feat = (float*)d_out;
    float* out_lbl  = out_feat + (size_t)N_TOTAL * NCLS;

    fps_kernel<<<1, 1024, 0, stream>>>(pos, col, min_d2, sample_idx, dxyz, drgb);
    normals_kernel<<<NSAMP / 256, 256, 0, stream>>>(dxyz, dnrm);

    const int prep_elems = NSAMP * 12 + 12 * HIDDEN + HIDDEN * 16 + 16;
    prep_kernel<<<(prep_elems + 255) / 256, 256, 0, stream>>>(
        dxyz, drgb, dnrm, W1, W3, b3, feats_pad, W1p, W3p, b3p);

    // layer1: (1024x12) @ (12x128)  -> relu -> h1      : 512 waves
    wmma_gemm_kernel<12, HIDDEN, 8, true><<<128, 128, 0, stream>>>(feats_pad, W1p, b1, h1);
    // layer2: (1024x128) @ (128x128)-> relu -> h2      : 512 waves
    wmma_gemm_kernel<HIDDEN, HIDDEN, 8, true><<<128, 128, 0, stream>>>(h1, W2, b2, h2);
    // layer3: (1024x128) @ (128x16) -> pred_pad        : 64 waves
    wmma_gemm_kernel<HIDDEN, 16, 1, false><<<16, 128, 0, stream>>>(h2, W3p, b3p, pred_pad);

    softmax_kernel<<<NSAMP / 256, 256, 0, stream>>>(pred_pad, feat_down, lbl_down);
    nearest_kernel<<<N_TOTAL / 256, 256, 0, stream>>>(pos, dxyz, feat_down, lbl_down,
                                                      out_feat, out_lbl);
}